// NeuralODE_51213190037643
// MI455X (gfx1250) — compile-verified
//
#include <hip/hip_runtime.h>

typedef float v2f __attribute__((ext_vector_type(2)));
typedef float v8f __attribute__((ext_vector_type(8)));

// lane^16 half-wave exchange on the VALU pipe: v_permlanex16_b32 with
// identity nibble selects (sel for lane i = i) swaps the two 16-lane halves.
__device__ __forceinline__ float swap16(float v) {
    int s = __float_as_int(v);
    int r = __builtin_amdgcn_permlanex16(s, s, 0x76543210, 0xFEDCBA98,
                                         false, false);
    return __int_as_float(r);
}

// single-instruction ReLU (avoids the canonicalize clang emits for fmax/fmed3)
__device__ __forceinline__ float relu1(float v) {
    float r;
    asm("v_max_num_f32 %0, 0, %1" : "=v"(r) : "v"(v));
    return r;
}

__device__ __forceinline__ v8f wmma4(v2f a, v2f b, v8f c) {
    // V_WMMA_F32_16X16X4_F32 : D(16x16,f32) = A(16x4,f32) x B(4x16,f32) + C
    return __builtin_amdgcn_wmma_f32_16x16x4_f32(false, a, false, b, (short)0, c,
                                                 false, false);
}

struct NetRegs {
    v2f a1c0, a1c1;   // W1^T (5x8 -> padded 16x8) as A operand, K-chunks {0..3},{4..7}
    v2f a2c0, a2c1;   // W2^T (8x3 -> padded 16x8) as A operand
    v8f c1, c2;       // bias rows baked into accumulator init (zero in upper half)
};

// One drift eval for 16 batch elements. Invariant: all per-lane values
// (x*, u*) are ZERO in lanes 16-31, and every WMMA D result has zero rows
// 8-15 (upper half-wave). Half-wave merges are therefore plain adds around
// a full-EXEC half-swap -- completely branchless.
__device__ __forceinline__ void drift(const NetRegs& N,
                                      float x0, float x1, float x2,
                                      float u0, float u1,
                                      float& k0, float& k1, float& k2)
{
    // layer 1: h^T(8x16) = W1^T * z^T + b1 ; z rows K0=x0 K1=x1 K2=x2 K3=u0 K4=u1
    v2f bA; bA.x = x0 + swap16(x2);   // rows {K0 | K2}
    bA.y = x1 + swap16(u0);           // rows {K1 | K3}
    v2f bB; bB.x = u1; bB.y = 0.0f;   // rows {K4 | K6=0}, {K5=0 | K7=0}
    v8f acc = N.c1;
    acc = wmma4(N.a1c0, bA, acc);
    acc = wmma4(N.a1c1, bB, acc);
    float h[8];
#pragma unroll
    for (int i = 0; i < 8; ++i) h[i] = relu1(acc[i]);
    // repack h^T (feature f in VGPR f, lanes 0-15; zeros in lanes 16-31)
    v2f p; p.x = h[0] + swap16(h[2]); p.y = h[1] + swap16(h[3]);
    v2f q; q.x = h[4] + swap16(h[6]); q.y = h[5] + swap16(h[7]);
    // layer 2: y^T(3x16) = W2^T * h^T + b2
    v8f acc2 = N.c2;
    acc2 = wmma4(N.a2c0, p, acc2);
    acc2 = wmma4(N.a2c1, q, acc2);
    k0 = acc2[0]; k1 = acc2[1]; k2 = acc2[2];  // valid lanes 0-15, zero 16-31
}

#define NTILE 2   // independent 16-element tiles per wave (ILP across WMMA chains)

__global__ void __launch_bounds__(256)
node_rk4(const float* __restrict__ x0in, const float* __restrict__ ci,
         const float* __restrict__ tspan,
         const float* __restrict__ W1, const float* __restrict__ b1v,
         const float* __restrict__ W2, const float* __restrict__ b2v,
         float* __restrict__ out, int B, int T)
{
    const int tid  = blockIdx.x * blockDim.x + threadIdx.x;
    const int lane = tid & 31;
    const int m    = lane & 15;
    const bool hi  = lane >= 16;
    const int wave = tid >> 5;

    // ---- loop-invariant WMMA operands ----
    NetRegs N;
    N.a1c0.x = (m < 8) ? W1[(hi ? 2 : 0) * 8 + m] : 0.f;   // A(M=m, K={0|2})
    N.a1c0.y = (m < 8) ? W1[(hi ? 3 : 1) * 8 + m] : 0.f;   // A(M=m, K={1|3})
    N.a1c1.x = (!hi && m < 8) ? W1[4 * 8 + m] : 0.f;       // K={4|6}
    N.a1c1.y = 0.f;                                        // K={5|7}
    N.a2c0.x = (m < 3) ? W2[(hi ? 2 : 0) * 3 + m] : 0.f;
    N.a2c0.y = (m < 3) ? W2[(hi ? 3 : 1) * 3 + m] : 0.f;
    N.a2c1.x = (m < 3) ? W2[(hi ? 6 : 4) * 3 + m] : 0.f;
    N.a2c1.y = (m < 3) ? W2[(hi ? 7 : 5) * 3 + m] : 0.f;
#pragma unroll
    for (int i = 0; i < 8; ++i) N.c1[i] = hi ? 0.f : b1v[i];
#pragma unroll
    for (int i = 0; i < 8; ++i) N.c2[i] = (!hi && i < 3) ? b2v[i] : 0.f;

    int b[NTILE]; bool valid[NTILE];
    float x0[NTILE], x1[NTILE], x2[NTILE];
    const float* up[NTILE];
#pragma unroll
    for (int j = 0; j < NTILE; ++j) {
        int bb = wave * (16 * NTILE) + j * 16 + m;
        valid[j] = (bb < B);
        if (!valid[j]) bb = B - 1;              // clamp addresses, EXEC stays full
        b[j]  = bb;
        x0[j] = hi ? 0.f : x0in[3 * bb + 0];    // upper half-wave carries zeros
        x1[j] = hi ? 0.f : x0in[3 * bb + 1];
        x2[j] = hi ? 0.f : x0in[3 * bb + 2];
        up[j] = ci + (size_t)bb * (size_t)T * 2u;
    }

    for (int t = 0; t < T - 1; ++t) {
        const float ts0 = tspan[t], ts1 = tspan[t + 1];
        const float dt  = ts1 - ts0;
        const float hdt = 0.5f * dt;
        const float s   = dt * (1.0f / 6.0f);

        float u0[NTILE], u1[NTILE];
#pragma unroll
        for (int j = 0; j < NTILE; ++j) {
            const float2 uu = *(const float2*)(up[j] + 2 * t);
            __builtin_prefetch(up[j] + 2 * t + 256, 0, 0);
            u0[j] = hi ? 0.f : uu.x;
            u1[j] = hi ? 0.f : uu.y;
        }

        float ka[NTILE][3], kb[NTILE][3], kc[NTILE][3], kd[NTILE][3];
#pragma unroll
        for (int j = 0; j < NTILE; ++j)
            drift(N, x0[j], x1[j], x2[j], u0[j], u1[j],
                  ka[j][0], ka[j][1], ka[j][2]);
#pragma unroll
        for (int j = 0; j < NTILE; ++j)
            drift(N, x0[j] + hdt * ka[j][0], x1[j] + hdt * ka[j][1],
                  x2[j] + hdt * ka[j][2], u0[j], u1[j],
                  kb[j][0], kb[j][1], kb[j][2]);
#pragma unroll
        for (int j = 0; j < NTILE; ++j)
            drift(N, x0[j] + hdt * kb[j][0], x1[j] + hdt * kb[j][1],
                  x2[j] + hdt * kb[j][2], u0[j], u1[j],
                  kc[j][0], kc[j][1], kc[j][2]);
#pragma unroll
        for (int j = 0; j < NTILE; ++j)
            drift(N, x0[j] + dt * kc[j][0], x1[j] + dt * kc[j][1],
                  x2[j] + dt * kc[j][2], u0[j], u1[j],
                  kd[j][0], kd[j][1], kd[j][2]);

#pragma unroll
        for (int j = 0; j < NTILE; ++j) {
            x0[j] += s * (ka[j][0] + 2.f * kb[j][0] + 2.f * kc[j][0] + kd[j][0]);
            x1[j] += s * (ka[j][1] + 2.f * kb[j][1] + 2.f * kc[j][1] + kd[j][1]);
            x2[j] += s * (ka[j][2] + 2.f * kb[j][2] + 2.f * kc[j][2] + kd[j][2]);
            if (valid[j] && !hi) {
                const size_t o = ((size_t)t * (size_t)B + (size_t)b[j]) * 3u;
                out[o] = x0[j]; out[o + 1] = x1[j]; out[o + 2] = x2[j];
            }
        }
    }
    // reference appends the last state again as row T-1
#pragma unroll
    for (int j = 0; j < NTILE; ++j) {
        if (valid[j] && !hi) {
            const size_t o = ((size_t)(T - 1) * (size_t)B + (size_t)b[j]) * 3u;
            out[o] = x0[j]; out[o + 1] = x1[j]; out[o + 2] = x2[j];
        }
    }
}

extern "C" void kernel_launch(void* const* d_in, const int* in_sizes, int n_in,
                              void* d_out, int out_size, void* d_ws, size_t ws_size,
                              hipStream_t stream)
{
    const float* x0 = (const float*)d_in[0];   // [B,3]
    const float* ci = (const float*)d_in[1];   // [B,T,2]
    const float* ts = (const float*)d_in[2];   // [T]
    const float* W1 = (const float*)d_in[3];   // [5,8]
    const float* b1 = (const float*)d_in[4];   // [8]
    const float* W2 = (const float*)d_in[5];   // [8,3]
    const float* b2 = (const float*)d_in[6];   // [3]
    float* out = (float*)d_out;                // [T,B,3]

    const int B = in_sizes[0] / 3;
    const int T = in_sizes[2];
    const int waves   = (B + 16 * NTILE - 1) / (16 * NTILE);
    const int threads = waves * 32;
    const int block   = 256;
    const int grid    = (threads + block - 1) / block;
    node_rk4<<<grid, block, 0, stream>>>(x0, ci, ts, W1, b1, W2, b2, out, B, T);
}